// ModelNew_25056839205357
// MI455X (gfx1250) — compile-verified
//
#include <hip/hip_runtime.h>

// ---------------------------------------------------------------------------
// Nystrom attention for MI455X (gfx1250, wave32, WMMA + async LDS loads).
// BH=32, N=8192, D=64, M=32 landmarks. Memory-bound (~256MB min traffic),
// so all matrix math uses full-precision V_WMMA_F32_16X16X4_F32, and tile
// staging uses GLOBAL_LOAD_ASYNC_TO_LDS_B128 (ASYNCcnt) where available.
// ---------------------------------------------------------------------------

#define BHN   32
#define NN    8192
#define DD    64
#define MM    32
#define TILE  128          // rows of Q/V/A handled per block in phases 2 & 4
#define NTILES (NN / TILE) // 64
#define QSTR  68           // padded LDS stride for 64-wide rows (16B-aligned rows, banks spread by 4*lane)
#define ASTR  36           // padded LDS stride for 32-wide rows (16B-aligned rows, banks spread by 36*lane)
#define EPSF  1e-6f

typedef float v2f __attribute__((ext_vector_type(2)));
typedef float v8f __attribute__((ext_vector_type(8)));
typedef int   v4i __attribute__((ext_vector_type(4)));

__device__ __forceinline__ v8f wmma_f32(v2f a, v2f b, v8f c) {
  // D = A(16x4, f32) * B(4x16, f32) + C(16x16, f32)
  return __builtin_amdgcn_wmma_f32_16x16x4_f32(
      /*neg_a=*/false, a, /*neg_b=*/false, b,
      /*c_mod=*/(short)0, c, /*reuse_a=*/false, /*reuse_b=*/false);
}

// ---- async global->LDS staging (CDNA5 ASYNCcnt path) ----------------------
#if __has_builtin(__builtin_amdgcn_global_load_async_to_lds_b128)
#define HAS_ASYNC_LDS 1
typedef __attribute__((address_space(1))) v4i ga_v4i;   // global b128 chunk
typedef __attribute__((address_space(3))) v4i ls_v4i;   // LDS b128 chunk
#else
#define HAS_ASYNC_LDS 0
#endif

// Copy 16 bytes global -> LDS. dst/src must be 16B aligned.
__device__ __forceinline__ void copy16(float* lds_dst, const float* gsrc) {
#if HAS_ASYNC_LDS
  __builtin_amdgcn_global_load_async_to_lds_b128(
      (ga_v4i*)(void*)gsrc, (ls_v4i*)(void*)lds_dst, /*offset=*/0, /*cpol=*/0);
#else
  float4 v = *(const float4*)gsrc;
  lds_dst[0] = v.x; lds_dst[1] = v.y; lds_dst[2] = v.z; lds_dst[3] = v.w;
#endif
}

// Drain outstanding async copies before the workgroup barrier.
__device__ __forceinline__ void copy_fence() {
#if HAS_ASYNC_LDS
#if __has_builtin(__builtin_amdgcn_s_wait_asynccnt)
  __builtin_amdgcn_s_wait_asynccnt(0);
#else
  asm volatile("s_wait_asynccnt 0" ::: "memory");
#endif
#endif
}

// --------------------------------------------------------------------------
// Zero the Y / y1 accumulators (contiguous region in ws).
// --------------------------------------------------------------------------
__global__ void nys_zero(float* p, int n) {
  int i = blockIdx.x * blockDim.x + threadIdx.x;
  if (i < n) p[i] = 0.0f;
}

// --------------------------------------------------------------------------
// Phase 1: per-bh landmark math.  Bm = softmax(QL KL^T / 8),
// Bs = 0.5(Bm+Bm^T)+eps*I, Binv = Bs^{-1} via Gauss-Jordan in LDS.
// --------------------------------------------------------------------------
__global__ __launch_bounds__(256) void nys_prep(const float* __restrict__ Q,
                                                const float* __restrict__ K,
                                                float* __restrict__ Binv) {
  __shared__ __align__(16) float ql[MM * QSTR];
  __shared__ __align__(16) float kl[MM * QSTR];
  __shared__ __align__(16) float s [MM * ASTR];
  __shared__ __align__(16) float aug[MM * 65];   // [Bs | I], 32 x 64, stride 65
  __shared__ float fac[MM];

  const int bh  = blockIdx.x;
  const int tid = threadIdx.x;

  // Gather landmark rows: idx[m] = m*(N-1)/(M-1)
  for (int e = tid; e < MM * DD; e += 256) {
    int m = e >> 6, d = e & 63;
    long src = (long)m * (NN - 1) / (MM - 1);
    ql[m * QSTR + d] = Q[((long)bh * NN + src) * DD + d];
    kl[m * QSTR + d] = K[((long)bh * NN + src) * DD + d];
  }
  __syncthreads();

  // S = QL KL^T * scale  (32x32, 1024 dot-64s)
  for (int e = tid; e < MM * MM; e += 256) {
    int i = e >> 5, j = e & 31;
    float acc = 0.f;
    for (int d = 0; d < DD; ++d) acc += ql[i * QSTR + d] * kl[j * QSTR + d];
    s[i * ASTR + j] = acc * 0.125f;
  }
  __syncthreads();

  // Row softmax -> Bm (in place)
  if (tid < MM) {
    float mx = -1e30f;
    for (int j = 0; j < MM; ++j) mx = fmaxf(mx, s[tid * ASTR + j]);
    float sum = 0.f;
    for (int j = 0; j < MM; ++j) {
      float e2 = __expf(s[tid * ASTR + j] - mx);
      s[tid * ASTR + j] = e2;
      sum += e2;
    }
    float inv = 1.f / sum;
    for (int j = 0; j < MM; ++j) s[tid * ASTR + j] *= inv;
  }
  __syncthreads();

  // aug = [ Bs | I ]
  for (int e = tid; e < MM * MM; e += 256) {
    int i = e >> 5, j = e & 31;
    float bs = 0.5f * (s[i * ASTR + j] + s[j * ASTR + i]) + ((i == j) ? EPSF : 0.f);
    aug[i * 65 + j]      = bs;
    aug[i * 65 + 32 + j] = (i == j) ? 1.f : 0.f;
  }
  __syncthreads();

  // Gauss-Jordan (no pivoting: Bs is symmetric, diagonally loaded)
  for (int p = 0; p < MM; ++p) {
    float invp = 1.f / aug[p * 65 + p];
    __syncthreads();
    for (int c = tid; c < 64; c += 256) aug[p * 65 + c] *= invp;
    __syncthreads();
    if (tid < MM) fac[tid] = aug[tid * 65 + p];
    __syncthreads();
    for (int e = tid; e < MM * 64; e += 256) {
      int r = e >> 6, c = e & 63;
      if (r != p) aug[r * 65 + c] -= fac[r] * aug[p * 65 + c];
    }
    __syncthreads();
  }

  // Binv = right half of aug
  for (int e = tid; e < MM * MM; e += 256)
    Binv[bh * MM * MM + e] = aug[(e >> 5) * 65 + 32 + (e & 31)];
}

// --------------------------------------------------------------------------
// Phase 2: per 128-row tile:
//   S = Qt KL^T /8 (WMMA f32)  ->  A = rowsoftmax(S) -> store A to ws
//   Ypart = A^T V (WMMA f32)   ->  atomic-accumulate into Y, y1.
// Q and V tiles are staged memory->LDS with async b128 copies.
// --------------------------------------------------------------------------
__global__ __launch_bounds__(256) void nys_phase2(const float* __restrict__ Q,
                                                  const float* __restrict__ K,
                                                  const float* __restrict__ V,
                                                  float* __restrict__ Aout,
                                                  float* __restrict__ Yacc,
                                                  float* __restrict__ y1acc) {
  __shared__ __align__(16) float qv_lds[TILE * QSTR];  // Q tile, later V tile
  __shared__ __align__(16) float kl_lds[MM * QSTR];
  __shared__ __align__(16) float a_lds [TILE * ASTR];  // S, then A

  const int bh   = blockIdx.x / NTILES;
  const int tile = blockIdx.x % NTILES;
  const int row0 = tile * TILE;
  const int tid  = threadIdx.x;
  const int lane = tid & 31, wave = tid >> 5;
  const int half = lane >> 4, lm = lane & 15;

  // Q tile -> LDS (async b128, 16B chunks)
  const float* Qt = Q + ((long)bh * NN + row0) * DD;
  for (int e = tid; e < TILE * DD / 4; e += 256) {
    int r = e >> 4, c4 = e & 15;
    copy16(&qv_lds[r * QSTR + c4 * 4], Qt + e * 4);
  }
  // Gather KL landmarks (scattered rows -> regular loads)
  for (int e = tid; e < MM * DD; e += 256) {
    int m = e >> 6, d = e & 63;
    long src = (long)m * (NN - 1) / (MM - 1);
    kl_lds[m * QSTR + d] = K[((long)bh * NN + src) * DD + d];
  }
  // Prefetch the V tile into cache while GEMM1 runs (global_prefetch_b8)
  for (int e = tid; e < TILE; e += 256)
    __builtin_prefetch(V + ((long)bh * NN + row0 + e) * DD, 0, 0);
  copy_fence();
  __syncthreads();

  // --- GEMM1: S = Q_tile * KL^T (128x32), wave w -> m-tile w, n-tiles {0,1}
  {
    const int mt = wave;
    v8f acc0 = {}; v8f acc1 = {};
    for (int i = 0; i < 16; ++i) {
      int k0 = 4 * i + 2 * half;
      v2f a;  a.x  = qv_lds[(mt * 16 + lm) * QSTR + k0];
              a.y  = qv_lds[(mt * 16 + lm) * QSTR + k0 + 1];
      v2f b0; b0.x = kl_lds[lm * QSTR + k0];         // B = KL^T: (k,n)=KL[n][k]
              b0.y = kl_lds[lm * QSTR + k0 + 1];
      v2f b1; b1.x = kl_lds[(16 + lm) * QSTR + k0];
              b1.y = kl_lds[(16 + lm) * QSTR + k0 + 1];
      acc0 = wmma_f32(a, b0, acc0);
      acc1 = wmma_f32(a, b1, acc1);
    }
    const float scale = 0.125f;
    for (int r = 0; r < 8; ++r) {
      int m = mt * 16 + r + half * 8;
      a_lds[m * ASTR + lm]      = acc0[r] * scale;
      a_lds[m * ASTR + 16 + lm] = acc1[r] * scale;
    }
  }
  __syncthreads();

  // Row softmax (one thread per row)
  if (tid < TILE) {
    float mx = -1e30f;
    for (int j = 0; j < MM; ++j) mx = fmaxf(mx, a_lds[tid * ASTR + j]);
    float sum = 0.f;
    for (int j = 0; j < MM; ++j) {
      float e2 = __expf(a_lds[tid * ASTR + j] - mx);
      a_lds[tid * ASTR + j] = e2;
      sum += e2;
    }
    float inv = 1.f / sum;
    for (int j = 0; j < MM; ++j) a_lds[tid * ASTR + j] *= inv;
  }
  __syncthreads();

  // V tile -> LDS (async; reuses Q buffer, all waves are past GEMM1's barrier)
  const float* Vt = V + ((long)bh * NN + row0) * DD;
  for (int e = tid; e < TILE * DD / 4; e += 256) {
    int r = e >> 4, c4 = e & 15;
    copy16(&qv_lds[r * QSTR + c4 * 4], Vt + e * 4);
  }

  // y1 partial: column sums of A tile (overlaps with async V copy)
  if (tid < MM) {
    float sum = 0.f;
    for (int r = 0; r < TILE; ++r) sum += a_lds[r * ASTR + tid];
    atomicAdd(&y1acc[bh * MM + tid], sum);
  }
  // Store A tile to workspace (read again in phase 4)
  for (int e = tid; e < TILE * MM; e += 256) {
    int r = e >> 5, c = e & 31;
    Aout[((long)bh * NN + row0 + r) * MM + c] = a_lds[r * ASTR + c];
  }
  copy_fence();
  __syncthreads();

  // --- GEMM2: Ypart = A^T (32x128) * V (128x64); wave -> (mt = w>>2, nt = w&3)
  {
    const int mt = wave >> 2;
    const int nt = wave & 3;
    v8f acc = {};
    for (int i = 0; i < 32; ++i) {
      int k0 = 4 * i + 2 * half;
      v2f a; a.x = a_lds[k0 * ASTR + mt * 16 + lm];        // A^T[m][k] = A[k][m]
             a.y = a_lds[(k0 + 1) * ASTR + mt * 16 + lm];
      v2f b; b.x = qv_lds[k0 * QSTR + nt * 16 + lm];       // V[k][n]
             b.y = qv_lds[(k0 + 1) * QSTR + nt * 16 + lm];
      acc = wmma_f32(a, b, acc);
    }
    for (int r = 0; r < 8; ++r) {
      int m = mt * 16 + r + half * 8;
      int n = nt * 16 + lm;
      atomicAdd(&Yacc[(bh * MM + m) * DD + n], acc[r]);
    }
  }
}

// --------------------------------------------------------------------------
// Phase 3: X = Binv * Y (32x64), x1 = Binv * y1 (32).  Tiny.
// --------------------------------------------------------------------------
__global__ __launch_bounds__(256) void nys_solve(const float* __restrict__ Binv,
                                                 const float* __restrict__ Y,
                                                 const float* __restrict__ y1,
                                                 float* __restrict__ X,
                                                 float* __restrict__ x1) {
  __shared__ __align__(16) float b [MM * ASTR];
  __shared__ __align__(16) float ys[MM * QSTR];
  const int bh = blockIdx.x, tid = threadIdx.x;
  for (int e = tid; e < MM * MM; e += 256)
    b[(e >> 5) * ASTR + (e & 31)] = Binv[bh * MM * MM + e];
  for (int e = tid; e < MM * DD; e += 256)
    ys[(e >> 6) * QSTR + (e & 63)] = Y[bh * MM * DD + e];
  __syncthreads();
  for (int e = tid; e < MM * DD; e += 256) {
    int i = e >> 6, d = e & 63;
    float acc = 0.f;
    for (int k = 0; k < MM; ++k) acc += b[i * ASTR + k] * ys[k * QSTR + d];
    X[bh * MM * DD + e] = acc;
  }
  if (tid < MM) {
    float acc = 0.f;
    for (int k = 0; k < MM; ++k) acc += b[tid * ASTR + k] * y1[bh * MM + k];
    x1[bh * MM + tid] = acc;
  }
}

// --------------------------------------------------------------------------
// Phase 4: out = (A @ X) / max(A @ x1, 1e-20)
// A tile and X staged with async b128 copies.
// --------------------------------------------------------------------------
__global__ __launch_bounds__(256) void nys_out(const float* __restrict__ Aflat,
                                               const float* __restrict__ X,
                                               const float* __restrict__ x1,
                                               float* __restrict__ out) {
  __shared__ __align__(16) float a_lds [TILE * ASTR];
  __shared__ __align__(16) float x_lds [MM * DD];  // [k][n]; lanes read along n
  __shared__ float x1_lds[MM];
  __shared__ float den   [TILE];

  const int bh   = blockIdx.x / NTILES;
  const int tile = blockIdx.x % NTILES;
  const int row0 = tile * TILE;
  const int tid  = threadIdx.x;
  const int lane = tid & 31, wave = tid >> 5;
  const int half = lane >> 4, lm = lane & 15;

  // X (2048 floats, contiguous) -> LDS async
  const float* Xb = X + (long)bh * MM * DD;
  for (int e = tid; e < MM * DD / 4; e += 256)
    copy16(&x_lds[e * 4], Xb + e * 4);
  // A tile (128x32) -> LDS async (rows are 128B in global, 144B in LDS)
  const float* Ab = Aflat + ((long)bh * NN + row0) * MM;
  for (int e = tid; e < TILE * MM / 4; e += 256) {
    int r = e >> 3, c4 = e & 7;
    copy16(&a_lds[r * ASTR + c4 * 4], Ab + e * 4);
  }
  if (tid < MM) x1_lds[tid] = x1[bh * MM + tid];
  copy_fence();
  __syncthreads();

  if (tid < TILE) {
    float acc = 0.f;
    for (int j = 0; j < MM; ++j) acc += a_lds[tid * ASTR + j] * x1_lds[j];
    den[tid] = fmaxf(acc, 1e-20f);
  }
  __syncthreads();

  // num = A (128x32) @ X (32x64); wave w -> m-tile w, n-tiles 0..3
  const int mt = wave;
  for (int nt = 0; nt < 4; ++nt) {
    v8f acc = {};
    for (int i = 0; i < 8; ++i) {
      int k0 = 4 * i + 2 * half;
      v2f a; a.x = a_lds[(mt * 16 + lm) * ASTR + k0];
             a.y = a_lds[(mt * 16 + lm) * ASTR + k0 + 1];
      v2f b; b.x = x_lds[k0 * DD + nt * 16 + lm];
             b.y = x_lds[(k0 + 1) * DD + nt * 16 + lm];
      acc = wmma_f32(a, b, acc);
    }
    for (int r = 0; r < 8; ++r) {
      int m = mt * 16 + r + half * 8;
      int n = nt * 16 + lm;
      out[((long)bh * NN + row0 + m) * DD + n] = acc[r] / den[m];
    }
  }
}

// --------------------------------------------------------------------------
extern "C" void kernel_launch(void* const* d_in, const int* in_sizes, int n_in,
                              void* d_out, int out_size, void* d_ws, size_t ws_size,
                              hipStream_t stream) {
  (void)in_sizes; (void)n_in; (void)out_size; (void)ws_size;
  const float* Q = (const float*)d_in[0];
  const float* K = (const float*)d_in[1];
  const float* V = (const float*)d_in[2];
  float* outp = (float*)d_out;

  // Workspace layout (floats): A | Binv | Y | y1 | X | x1   (~34 MB)
  float* ws   = (float*)d_ws;
  float* A    = ws;                                   // BH*N*M = 8388608
  float* Binv = A    + (size_t)BHN * NN * MM;         // BH*M*M = 32768
  float* Y    = Binv + (size_t)BHN * MM * MM;         // BH*M*D = 65536
  float* y1   = Y    + (size_t)BHN * MM * DD;         // BH*M   = 1024
  float* X    = y1   + (size_t)BHN * MM;              // BH*M*D = 65536
  float* x1   = X    + (size_t)BHN * MM * DD;         // BH*M   = 1024

  const int zn = BHN * MM * DD + BHN * MM;            // Y and y1 contiguous
  nys_zero  <<<(zn + 255) / 256, 256, 0, stream>>>(Y, zn);
  nys_prep  <<<BHN,              256, 0, stream>>>(Q, K, Binv);
  nys_phase2<<<BHN * NTILES,     256, 0, stream>>>(Q, K, V, A, Y, y1);
  nys_solve <<<BHN,              256, 0, stream>>>(Binv, Y, y1, X, x1);
  nys_out   <<<BHN * NTILES,     256, 0, stream>>>(A, X, x1, outp);
}